// RESCALModel_91233695301845
// MI455X (gfx1250) — compile-verified
//
#include <hip/hip_runtime.h>
#include <hip/hip_bf16.h>
#include <stddef.h>

// ---------------------------------------------------------------------------
// RESCAL bilinear scoring on MI455X (gfx1250, wave32, WMMA).
//
// score_b = h_b^T R_{rel(b)} t_b  (and same with negative samples).
// Only 200 distinct R matrices (12.8 MB total -> L2 resident). We bin the
// 16384 batch items by relation; every 16-item chunk with a shared R is a
// real GEMM  V = R (128x128) x T (128x16)  computed with
// V_WMMA_F32_16X16X4_F32 (fp32, matching the reference dtype), followed by a
// deterministic column reduction against H.
// ---------------------------------------------------------------------------

#define DIM 128
#define RPITCH 132   // LDS row pitch for R (floats): bank-conflict-free b64 reads

typedef __attribute__((ext_vector_type(2))) float v2f;
typedef __attribute__((ext_vector_type(8))) float v8f;

// -------------------- binning kernels --------------------

__global__ void k_init(int* counts, int* chunk_rel, int* bin_items,
                       int nrel, int maxChunks) {
  int i = blockIdx.x * blockDim.x + threadIdx.x;
  if (i < nrel) counts[i] = 0;
  if (i < maxChunks) chunk_rel[i] = -1;
  if (i < maxChunks * 16) bin_items[i] = -1;
}

__global__ void k_hist(const int* __restrict__ rel, int* counts, int B) {
  int i = blockIdx.x * blockDim.x + threadIdx.x;
  if (i < B) atomicAdd(&counts[rel[i]], 1);
}

// tiny serial pass over 200 relations: build chunk->relation map and bin cursors
__global__ void k_scan(const int* __restrict__ counts, int* cursor,
                       int* chunk_rel, int nrel) {
  if (blockIdx.x == 0 && threadIdx.x == 0) {
    int c = 0;
    for (int r = 0; r < nrel; ++r) {
      int nch = (counts[r] + 15) >> 4;
      cursor[r] = c * 16;                    // first slot of this relation's bin
      for (int j = 0; j < nch; ++j) chunk_rel[c + j] = r;
      c += nch;
    }
  }
}

__global__ void k_scatter(const int* __restrict__ rel, int* cursor,
                          int* bin_items, int B) {
  int i = blockIdx.x * blockDim.x + threadIdx.x;
  if (i < B) {
    int p = atomicAdd(&cursor[rel[i]], 1);
    bin_items[p] = i;
  }
}

// -------------------- compute kernel --------------------
// One 256-thread workgroup (8 wave32) per 16-item chunk sharing one R.
// Wave w owns output rows [16w, 16w+16) of V = R x T (and of R x NT).

__global__ __launch_bounds__(256) void k_rescal_chunks(
    const int* __restrict__ bin_items, const int* __restrict__ chunk_rel,
    const int* __restrict__ heads, const int* __restrict__ tails,
    const int* __restrict__ nheads, const int* __restrict__ ntails,
    const float* __restrict__ ent, const float* __restrict__ rmats,
    float* __restrict__ out, int B) {
  __shared__ float Rlds[DIM * RPITCH];   // 67584 B, padded rows
  __shared__ float Tp[64 * 32];          // t  vectors, [k/2][n][2] interleave
  __shared__ float NTp[64 * 32];         // nt vectors, same layout
  __shared__ float Hl[DIM * 16];         // h  vectors, [d][n]
  __shared__ float NHl[DIM * 16];        // nh vectors, [d][n]
  __shared__ float partP[256];
  __shared__ float partN[256];

  const int c = blockIdx.x;
  const int r = chunk_rel[c];
  if (r < 0) return;                     // padding chunk (uniform exit)
  const int tid = threadIdx.x;

  // ---- stage R (64 KB) into LDS: coalesced b128 loads, padded-row stores ----
  const float4* Rsrc = (const float4*)(rmats + (size_t)r * DIM * DIM);
  for (int i = tid; i < (DIM * DIM) / 4; i += 256) {
    float4 v = Rsrc[i];
    int e = i * 4;
    int row = e >> 7, col = e & 127;
    *(float4*)&Rlds[row * RPITCH + col] = v;
  }

  // ---- gather the 4 embedding rows per slot (coalesced: 128 lanes per row) ----
  for (int it = 0; it < 8; ++it) {
    int slot = (tid >> 7) + it * 2;      // 0..15
    int k = tid & 127;
    int b = bin_items[c * 16 + slot];
    float hv = 0.f, tv = 0.f, nhv = 0.f, ntv = 0.f;
    if (b >= 0) {
      hv  = ent[(size_t)heads[b]  * DIM + k];
      tv  = ent[(size_t)tails[b]  * DIM + k];
      nhv = ent[(size_t)nheads[b] * DIM + k];
      ntv = ent[(size_t)ntails[b] * DIM + k];
    }
    int pi = (k >> 1) * 32 + slot * 2 + (k & 1);   // K-pair interleave
    Tp[pi] = tv;
    NTp[pi] = ntv;
    Hl[k * 16 + slot] = hv;
    NHl[k * 16 + slot] = nhv;
  }
  __syncthreads();

  // ---- WMMA: V = R x T and VN = R x NT, K=4 steps, fp32 accumulate ----
  const int wave = tid >> 5;             // 0..7
  const int lane = tid & 31;
  const int n = lane & 15;               // column / item slot
  const int hi = lane >> 4;              // half-wave selects K sub-pair
  const int arow = wave * 16 + n;        // A-matrix row for this lane

  v8f cp = {};
  v8f cn = {};
  for (int k0 = 0; k0 < DIM; k0 += 4) {
    int kb = k0 + hi * 2;
    v2f a  = *(const v2f*)&Rlds[arow * RPITCH + kb];       // A[16x4] tile
    v2f bp = *(const v2f*)&Tp[(kb >> 1) * 32 + n * 2];     // B[4x16] tile (pos)
    v2f bn = *(const v2f*)&NTp[(kb >> 1) * 32 + n * 2];    // B[4x16] tile (neg)
    cp = __builtin_amdgcn_wmma_f32_16x16x4_f32(false, a, false, bp,
                                               (short)0, cp, false, false);
    cn = __builtin_amdgcn_wmma_f32_16x16x4_f32(false, a, false, bn,
                                               (short)0, cn, false, false);
  }

  // ---- deterministic reduction: score_n = sum_d H[d][n] * V[d][n] ----
  float pP = 0.f, pN = 0.f;
  for (int v = 0; v < 8; ++v) {
    int d = wave * 16 + hi * 8 + v;      // accumulator row -> absolute d
    pP += cp[v] * Hl[d * 16 + n];
    pN += cn[v] * NHl[d * 16 + n];
  }
  partP[tid] = pP;
  partN[tid] = pN;
  __syncthreads();

  if (tid < 16) {
    float sp = 0.f, sn = 0.f;
    for (int w2 = 0; w2 < 8; ++w2) {
      sp += partP[w2 * 32 + tid] + partP[w2 * 32 + 16 + tid];
      sn += partN[w2 * 32 + tid] + partN[w2 * 32 + 16 + tid];
    }
    int b = bin_items[c * 16 + tid];
    if (b >= 0) {
      out[b] = sp;          // positive scores: out[0..B)
      out[B + b] = sn;      // negative scores: out[B..2B)
    }
  }
}

// -------------------- launch --------------------

extern "C" void kernel_launch(void* const* d_in, const int* in_sizes, int n_in,
                              void* d_out, int out_size, void* d_ws, size_t ws_size,
                              hipStream_t stream) {
  const int* heads  = (const int*)d_in[0];
  const int* tails  = (const int*)d_in[1];
  const int* nheads = (const int*)d_in[2];
  const int* ntails = (const int*)d_in[3];
  const int* rel    = (const int*)d_in[4];
  const float* ent  = (const float*)d_in[5];
  const float* rmat = (const float*)d_in[6];
  float* out = (float*)d_out;

  const int B = in_sizes[0];
  const int nrel = in_sizes[6] / (DIM * DIM);
  const int maxChunks = (B + 15) / 16 + nrel;   // worst-case padded chunk count

  // scratch layout (ints): counts | cursor | chunk_rel | bin_items
  int* counts    = (int*)d_ws;
  int* cursor    = counts + nrel;
  int* chunk_rel = cursor + nrel;
  int* bin_items = chunk_rel + maxChunks;
  (void)ws_size;

  const int initN = maxChunks * 16;     // dominates nrel and maxChunks
  k_init<<<(initN + 255) / 256, 256, 0, stream>>>(counts, chunk_rel, bin_items,
                                                  nrel, maxChunks);
  k_hist<<<(B + 255) / 256, 256, 0, stream>>>(rel, counts, B);
  k_scan<<<1, 1, 0, stream>>>(counts, cursor, chunk_rel, nrel);
  k_scatter<<<(B + 255) / 256, 256, 0, stream>>>(rel, cursor, bin_items, B);
  k_rescal_chunks<<<maxChunks, 256, 0, stream>>>(
      bin_items, chunk_rel, heads, tails, nheads, ntails, ent, rmat, out, B);
}